// Decoder_85014582657181
// MI455X (gfx1250) — compile-verified
//
#include <hip/hip_runtime.h>
#include <math.h>

typedef __attribute__((ext_vector_type(2))) float v2f;
typedef __attribute__((ext_vector_type(8))) float v8f;

#define BB   2048
#define HH   2048
#define MMIX 20
#define KX   133
#define KXP  136               // padded to multiple of 4
#define KTOT (KXP + HH)        // 2184 = 34*64 + 8
#define LSTR 67                // LDS row stride -> conflict-free fragment reads

// d_out layout (concatenated reference outputs)
#define O_PI   (BB * 5)
#define O_MUX  (O_PI  + BB * MMIX)
#define O_MUY  (O_MUX + BB * MMIX)
#define O_SDX  (O_MUY + BB * MMIX)
#define O_SDY  (O_SDX + BB * MMIX)
#define O_RHO  (O_SDY + BB * MMIX)
#define O_Q    (O_RHO + BB * MMIX)

__device__ __forceinline__ float sigmoidf_(float x) {
  return 1.0f / (1.0f + __expf(-x));
}

// ---------------------------------------------------------------------------
// Kernel 0: build padded X = [z | stroke | 0] (B x 136), padded W_ih
// (8192 x 136, zero cols 133..135), padded W_proj (128 x 2048, zero rows
// 123..127).
// ---------------------------------------------------------------------------
__global__ void prep_kernel(const float* __restrict__ z,
                            const float* __restrict__ stroke,
                            const float* __restrict__ wih,
                            const float* __restrict__ wproj,
                            float* __restrict__ xpad,
                            float* __restrict__ wihp,
                            float* __restrict__ wpp) {
  const int stride = gridDim.x * blockDim.x;
  const int t0 = blockIdx.x * blockDim.x + threadIdx.x;
  for (int i = t0; i < BB * KXP; i += stride) {
    const int r = i / KXP, c = i - r * KXP;
    float v = 0.f;
    if (c < 128)      v = z[r * 128 + c];
    else if (c < KX)  v = stroke[r * 5 + (c - 128)];
    xpad[i] = v;
  }
  for (int i = t0; i < 4 * HH * KXP; i += stride) {
    const int r = i / KXP, c = i - r * KXP;
    wihp[i] = (c < KX) ? wih[r * KX + c] : 0.f;
  }
  for (int i = t0; i < 128 * HH; i += stride) {
    const int r = i / HH;
    wpp[i] = (r < 123) ? wproj[i] : 0.f;
  }
}

// ---------------------------------------------------------------------------
// Kernel 1: fused gates GEMM (2048 x 8192 x 2184 fp32, v_wmma_f32_16x16x4)
// + LSTM cell update. Each wave owns a 64(batch) x 16(h) tile for ALL four
// gates: 16 v8f accumulators. Per K-step: 4 LDS A-fragments + 4 global
// B-fragments feed 16 WMMAs, so weight (L2) bytes and address VALU amortize
// 4x vs a 16x16 tile. The 67MB gates matrix is never materialized.
// ---------------------------------------------------------------------------
__global__ __launch_bounds__(256) void gates_lstm_kernel(
    const float* __restrict__ xpad,   // B x 136
    const float* __restrict__ hidden, // B x 2048
    const float* __restrict__ wihp,   // 8192 x 136 (padded)
    const float* __restrict__ whh,    // 8192 x 2048
    const float* __restrict__ bih,    // 8192
    const float* __restrict__ bhh,    // 8192
    const float* __restrict__ cell,   // B x 2048
    float* __restrict__ hnew)         // B x 2048
{
  __shared__ float lds[64 * LSTR];   // 64-row A strip panel, 17 KB
  const int tid  = threadIdx.x;
  const int wave = tid >> 5;
  const int lane = tid & 31;
  const int lrow = lane & 15;
  const int lhi  = lane >> 4;
  const int m0 = blockIdx.y * 64;               // batch tile (64 rows)
  const int n0 = blockIdx.x * 128 + wave * 16;  // hidden-unit tile

  v8f acc[4][4];                                // [gate][m-subtile]
#pragma unroll
  for (int g = 0; g < 4; ++g)
#pragma unroll
    for (int s = 0; s < 4; ++s) acc[g][s] = (v8f){0.f,0.f,0.f,0.f,0.f,0.f,0.f,0.f};

  const int a_base = lrow * LSTR + 2 * lhi;     // A-fragment LDS base (subtile 0)
  const int brow   = n0 + lrow;                 // B-fragment W row per lane

  for (int p = 0; p < 35; ++p) {
    const int kb = p * 64;
    const int pw = (p < 34) ? 64 : (KTOT - 34 * 64); // 64 or 8
    const int sh = (p < 34) ? 6 : 3;
    __syncthreads();
    // stage A strip panel: rows m0..m0+63, K-cols kb..kb+pw-1 (coalesced)
    for (int e = tid; e < 64 * pw; e += 256) {
      const int r = e >> sh, c = e & (pw - 1);
      const int gk = kb + c;
      float v;
      if (gk < KXP) v = xpad[(m0 + r) * KXP + gk];
      else          v = hidden[(m0 + r) * HH + (gk - KXP)];
      lds[r * LSTR + c] = v;
    }
    __syncthreads();
    for (int kk = 0; kk < pw; kk += 4) {
      const int gk = kb + kk;                   // wave-uniform
      const float* wsrc;
      int wstride, wcol;
      if (gk < KXP) { wsrc = wihp; wstride = KXP; wcol = gk; }
      else          { wsrc = whh;  wstride = HH;  wcol = gk - KXP; }
      wcol += 2 * lhi;
      // 4 B-fragments (one per gate)
      v2f bf[4];
#pragma unroll
      for (int g = 0; g < 4; ++g) {
        const float* wr = wsrc + (size_t)(g * HH + brow) * wstride + wcol;
        bf[g].x = wr[0];
        bf[g].y = wr[1];
      }
      // 4 A-fragments (one per 16-row batch subtile)
      v2f af[4];
#pragma unroll
      for (int s = 0; s < 4; ++s) {
        const int ab = a_base + s * 16 * LSTR + kk;
        af[s].x = lds[ab];
        af[s].y = lds[ab + 1];
      }
      // 16 WMMAs
#pragma unroll
      for (int g = 0; g < 4; ++g)
#pragma unroll
        for (int s = 0; s < 4; ++s)
          acc[g][s] = __builtin_amdgcn_wmma_f32_16x16x4_f32(
              false, af[s], false, bf[g], (short)0, acc[g][s], false, false);
    }
  }

  // Epilogue: bias + LSTM nonlinearity in-register, write h_new only.
  const int n = n0 + lrow;
  const float biI = bih[0 * HH + n] + bhh[0 * HH + n];
  const float biF = bih[1 * HH + n] + bhh[1 * HH + n];
  const float biG = bih[2 * HH + n] + bhh[2 * HH + n];
  const float biO = bih[3 * HH + n] + bhh[3 * HH + n];
#pragma unroll
  for (int s = 0; s < 4; ++s) {
#pragma unroll
    for (int r = 0; r < 8; ++r) {
      const int m = m0 + s * 16 + r + 8 * lhi;
      const float ig = sigmoidf_(acc[0][s][r] + biI);
      const float fg = sigmoidf_(acc[1][s][r] + biF);
      const float gg = tanhf(acc[2][s][r] + biG);
      const float og = sigmoidf_(acc[3][s][r] + biO);
      const float cn = fg * cell[m * HH + n] + ig * gg;
      hnew[m * HH + n] = og * tanhf(cn);
    }
  }
}

// ---------------------------------------------------------------------------
// Kernel 2: out = h_new @ W_proj^T + b_proj  (2048 x 128pad x 2048, WMMA).
// 8 waves of a block cover all 8 N-tiles and share the A strip via LDS.
// ---------------------------------------------------------------------------
__global__ __launch_bounds__(256) void proj_kernel(
    const float* __restrict__ hnew,  // B x 2048
    const float* __restrict__ wpp,   // 128 x 2048 (padded)
    const float* __restrict__ bproj, // 123
    float* __restrict__ outp)        // B x 128
{
  __shared__ float lds[16 * LSTR];
  const int tid  = threadIdx.x;
  const int wave = tid >> 5;
  const int lane = tid & 31;
  const int lrow = lane & 15;
  const int lhi  = lane >> 4;
  const int m0 = blockIdx.x * 16;
  const int n0 = wave * 16;

  v8f acc = {};
  const int a_base = lrow * LSTR + 2 * lhi;
  const int brow   = n0 + lrow;

  for (int p = 0; p < HH / 64; ++p) {
    const int kb = p * 64;
    __syncthreads();
    for (int e = tid; e < 16 * 64; e += 256) {
      const int r = e >> 6, c = e & 63;
      lds[r * LSTR + c] = hnew[(m0 + r) * HH + kb + c];
    }
    __syncthreads();
    for (int kk = 0; kk < 64; kk += 4) {
      v2f a;
      a.x = lds[a_base + kk];
      a.y = lds[a_base + kk + 1];
      const float* wr = wpp + (size_t)brow * HH + kb + kk + 2 * lhi;
      v2f bv; bv.x = wr[0]; bv.y = wr[1];
      acc = __builtin_amdgcn_wmma_f32_16x16x4_f32(false, a, false, bv, (short)0, acc, false, false);
    }
  }

  const int n = n0 + lrow;
  const float bias = (n < 123) ? bproj[n] : 0.f;
#pragma unroll
  for (int r = 0; r < 8; ++r) {
    const int m = m0 + r + 8 * lhi;
    outp[m * 128 + n] = acc[r] + bias;
  }
}

// ---------------------------------------------------------------------------
// Kernel 3: GMM head — softmax over 20 mixtures, sampling, pen one-hot.
// One thread per batch row (negligible cost).
// ---------------------------------------------------------------------------
__global__ void sample_kernel(const float* __restrict__ outp,
                              const float* __restrict__ n1,
                              const float* __restrict__ n2,
                              const float* __restrict__ uc,
                              const float* __restrict__ up,
                              float* __restrict__ out) {
  const int b = blockIdx.x * blockDim.x + threadIdx.x;
  if (b >= BB) return;
  const float* o = outp + b * 128;

  float mx = -3.4e38f;
  for (int m = 0; m < MMIX; ++m) mx = fmaxf(mx, o[6 * m]);
  float s = 0.f;
  for (int m = 0; m < MMIX; ++m) s += __expf(o[6 * m] - mx);
  const float inv = 1.f / s;

  const float u = uc[b];
  float cum = 0.f;
  int cnt = 0;
  for (int m = 0; m < MMIX; ++m) {
    const float p = __expf(o[6 * m] - mx) * inv;
    out[O_PI  + b * MMIX + m] = p;
    out[O_MUX + b * MMIX + m] = o[6 * m + 1];
    out[O_MUY + b * MMIX + m] = o[6 * m + 2];
    out[O_SDX + b * MMIX + m] = __expf(o[6 * m + 3]);
    out[O_SDY + b * MMIX + m] = __expf(o[6 * m + 4]);
    out[O_RHO + b * MMIX + m] = tanhf(o[6 * m + 5]);
    cum += p;
    cnt += (cum < u) ? 1 : 0;
  }
  const int comp = cnt > (MMIX - 1) ? (MMIX - 1) : cnt;
  const float mxv = o[6 * comp + 1];
  const float myv = o[6 * comp + 2];
  const float sxv = __expf(o[6 * comp + 3]);
  const float syv = __expf(o[6 * comp + 4]);
  const float rv  = tanhf(o[6 * comp + 5]);
  const float a1 = n1[b], a2 = n2[b];
  const float xs = mxv + sxv * a1;
  const float ys = myv + syv * (rv * a1 + sqrtf(fmaxf(1.f - rv * rv, 0.f)) * a2);

  const float q0l = o[120], q1l = o[121], q2l = o[122];
  const float qm = fmaxf(q0l, fmaxf(q1l, q2l));
  const float e0 = __expf(q0l - qm), e1 = __expf(q1l - qm), e2 = __expf(q2l - qm);
  const float qs = 1.f / (e0 + e1 + e2);
  const float q0 = e0 * qs, q1 = e1 * qs, q2 = e2 * qs;
  out[O_Q + b * 3 + 0] = q0;
  out[O_Q + b * 3 + 1] = q1;
  out[O_Q + b * 3 + 2] = q2;

  const float uq = up[b];
  int pcnt = ((q0 < uq) ? 1 : 0) + ((q0 + q1 < uq) ? 1 : 0) + ((q0 + q1 + q2 < uq) ? 1 : 0);
  const int pen = pcnt > 2 ? 2 : pcnt;

  float* sn = out + b * 5;
  sn[0] = xs;
  sn[1] = ys;
  sn[2] = (pen == 0) ? 1.f : 0.f;
  sn[3] = (pen == 1) ? 1.f : 0.f;
  sn[4] = (pen == 2) ? 1.f : 0.f;
}

// ---------------------------------------------------------------------------
extern "C" void kernel_launch(void* const* d_in, const int* in_sizes, int n_in,
                              void* d_out, int out_size, void* d_ws, size_t ws_size,
                              hipStream_t stream) {
  (void)in_sizes; (void)n_in; (void)out_size; (void)ws_size;
  const float* z      = (const float*)d_in[0];
  const float* stroke = (const float*)d_in[1];
  const float* hidden = (const float*)d_in[2];
  const float* cell   = (const float*)d_in[3];
  const float* W_ih   = (const float*)d_in[4];
  const float* W_hh   = (const float*)d_in[5];
  const float* b_ih   = (const float*)d_in[6];
  const float* b_hh   = (const float*)d_in[7];
  const float* W_proj = (const float*)d_in[8];
  const float* b_proj = (const float*)d_in[9];
  const float* n1     = (const float*)d_in[10];
  const float* n2     = (const float*)d_in[11];
  const float* u_comp = (const float*)d_in[12];
  const float* u_pen  = (const float*)d_in[13];

  // Workspace layout (floats): ~24.4 MB total
  float* ws   = (float*)d_ws;
  float* xpad = ws;                              // 2048*136   = 278528
  float* wihp = xpad + (size_t)BB * KXP;         // 8192*136   = 1114112
  float* hnew = wihp + (size_t)4 * HH * KXP;     // 2048*2048  = 4194304
  float* wpp  = hnew + (size_t)BB * HH;          // 128*2048   = 262144
  float* outp = wpp  + (size_t)128 * HH;         // 2048*128   = 262144

  prep_kernel<<<1024, 256, 0, stream>>>(z, stroke, W_ih, W_proj, xpad, wihp, wpp);
  gates_lstm_kernel<<<dim3(HH / 128, BB / 64), 256, 0, stream>>>(
      xpad, hidden, wihp, W_hh, b_ih, b_hh, cell, hnew);
  proj_kernel<<<dim3(BB / 16), 256, 0, stream>>>(hnew, wpp, b_proj, outp);
  sample_kernel<<<BB / 256, 256, 0, stream>>>(outp, n1, n2, u_comp, u_pen, (float*)d_out);
}